// MANO_SMPL_23536420782296
// MI455X (gfx1250) — compile-verified
//
#include <hip/hip_runtime.h>
#include <hip/hip_bf16.h>

typedef float v2f __attribute__((ext_vector_type(2)));
typedef float v8f __attribute__((ext_vector_type(8)));

#define N_VERTS 778
#define VPAD    780      // K padded to multiple of 4 for f32 WMMA (K=4)
#define NJ_ALL  21
#define NJ_CH   16
#define KDIM    145      // 10 (beta) + 135 (pose feature)
#define KPAD    148      // padded feature dim (multiple of 4)
#define DROWS   160      // Dirs rows padded to 10 WMMA M-tiles
#define NK2     336      // 21*16 (jj,j) pairs
#define NCOLS   1008     // 336*3 real columns
#define NCP     1024     // padded to 64 exact 16-wide N tiles

// ---------------------------------------------------------------------------
__global__ void k_zero(float* __restrict__ p, int n) {
    int i = blockIdx.x * blockDim.x + threadIdx.x;
    if (i < n) p[i] = 0.0f;
}

// ---------------------------------------------------------------------------
// Rodrigues (matches reference: angle = ||r + 1e-8||, axis = r/angle)
// ---------------------------------------------------------------------------
__device__ __forceinline__ void rodrigues9(float rx, float ry, float rz, float* R) {
    float ax = rx + 1e-8f, ay = ry + 1e-8f, az = rz + 1e-8f;
    float angle = sqrtf(ax * ax + ay * ay + az * az);
    float inv = 1.0f / angle;
    float x = rx * inv, y = ry * inv, z = rz * inv;
    float c = cosf(angle), s = sinf(angle), t = 1.0f - c;
    R[0] = 1.0f - t * (y * y + z * z);
    R[1] = -s * z + t * (x * y);
    R[2] =  s * y + t * (x * z);
    R[3] =  s * z + t * (x * y);
    R[4] = 1.0f - t * (x * x + z * z);
    R[5] = -s * x + t * (y * z);
    R[6] = -s * y + t * (x * z);
    R[7] =  s * x + t * (y * z);
    R[8] = 1.0f - t * (x * x + y * y);
}

// ---------------------------------------------------------------------------
// SJ[d][j*3+c] = sum_v shapedirs[d, v*3+c] * Jr[v,j]  (j<16)
// Jt[j*3+c]    = sum_v v_template[v*3+c]   * Jr[v,j]
// ---------------------------------------------------------------------------
__global__ void k_SJ(const float* __restrict__ shapedirs, const float* __restrict__ v_template,
                     const float* __restrict__ Jreg, float* __restrict__ SJ, float* __restrict__ Jt) {
    int t = blockIdx.x * blockDim.x + threadIdx.x;
    if (t >= 528) return;
    if (t < 480) {
        int d = t / 48, col = t % 48, j = col / 3, c = col % 3;
        float acc = 0.0f;
        for (int v = 0; v < N_VERTS; v++)
            acc += shapedirs[(size_t)d * 2334 + v * 3 + c] * Jreg[v * NJ_ALL + j];
        SJ[t] = acc;
    } else {
        int col = t - 480, j = col / 3, c = col % 3;
        float acc = 0.0f;
        for (int v = 0; v < N_VERTS; v++)
            acc += v_template[v * 3 + c] * Jreg[v * NJ_ALL + j];
        Jt[col] = acc;
    }
}

// ---------------------------------------------------------------------------
// W2T[k][v] = Jr[v, k/16] * w[v, k%16],  v >= 778 -> 0   (transposed + padded)
// ---------------------------------------------------------------------------
__global__ void k_W2T(const float* __restrict__ Jreg, const float* __restrict__ weights,
                      float* __restrict__ W2T) {
    int idx = blockIdx.x * blockDim.x + threadIdx.x;
    if (idx >= NK2 * VPAD) return;
    int k = idx / VPAD, v = idx % VPAD;
    W2T[idx] = (v < N_VERTS) ? Jreg[v * NJ_ALL + k / 16] * weights[v * NJ_CH + (k % 16)] : 0.0f;
}

// ---------------------------------------------------------------------------
// DqPad[q][d][v] = Dirs[d, v*3+q]  (d<145, v<778; else 0)
// Dirs row d: d<10 -> shapedirs[d], else posedirs[d-10]
// ---------------------------------------------------------------------------
__global__ void k_Dq(const float* __restrict__ shapedirs, const float* __restrict__ posedirs,
                     float* __restrict__ DqPad) {
    int idx = blockIdx.x * blockDim.x + threadIdx.x;
    if (idx >= 3 * DROWS * VPAD) return;
    int q = idx / (DROWS * VPAD);
    int rem = idx % (DROWS * VPAD);
    int d = rem / VPAD, v = rem % VPAD;
    float val = 0.0f;
    if (d < KDIM && v < N_VERTS) {
        const float* Dr = (d < 10) ? (shapedirs + (size_t)d * 2334)
                                   : (posedirs + (size_t)(d - 10) * 2334);
        val = Dr[v * 3 + q];
    }
    DqPad[idx] = val;
}

// ---------------------------------------------------------------------------
// biasC[col] (col<1008): sum_v W2T[k][v]*v_template[v,q];  cols 1008..1023 = 0
// M3[k] = sum_v W2T[k][v]
// ---------------------------------------------------------------------------
__global__ void k_bias(const float* __restrict__ W2T, const float* __restrict__ v_template,
                       float* __restrict__ biasC, float* __restrict__ M3) {
    int idx = blockIdx.x * blockDim.x + threadIdx.x;
    if (idx >= NCP + NK2) return;
    if (idx < NCP) {
        float acc = 0.0f;
        if (idx < NCOLS) {
            int k = idx / 3, q = idx % 3;
            const float* w = W2T + (size_t)k * VPAD;
            for (int v = 0; v < N_VERTS; v++) acc += w[v] * v_template[v * 3 + q];
        }
        biasC[idx] = acc;
    } else {
        int k = idx - NCP;
        const float* w = W2T + (size_t)k * VPAD;
        float acc = 0.0f;
        for (int v = 0; v < N_VERTS; v++) acc += w[v];
        M3[k] = acc;
    }
}

// ---------------------------------------------------------------------------
// E2T[(k*3+q)][d] = sum_v DqPad[q][d][v] * W2T[k][v]
// WMMA f32 16x16x4; grid (10, 21, 3), block 32 (one wave per 16x16 tile).
// Branch-free inner loop: both fragments are contiguous aligned b64 loads.
// ---------------------------------------------------------------------------
__global__ void k_E2(const float* __restrict__ DqPad, const float* __restrict__ W2T,
                     float* __restrict__ E2T) {
    const int lane = threadIdx.x;
    const int half = lane >> 4, l15 = lane & 15;
    const int dm0 = blockIdx.x * 16;
    const int kn0 = blockIdx.y * 16;
    const int q   = blockIdx.z;

    const float* ap = DqPad + ((size_t)q * DROWS + (dm0 + l15)) * VPAD;   // A row (d)
    const float* bp = W2T + (size_t)(kn0 + l15) * VPAD;                   // B col (k)

    v8f acc = {0.f, 0.f, 0.f, 0.f, 0.f, 0.f, 0.f, 0.f};
    for (int vv = 0; vv < VPAD / 4; vv++) {
        int kb = vv * 4 + half * 2;
        v2f a = *(const v2f*)(ap + kb);
        v2f b = *(const v2f*)(bp + kb);
        acc = __builtin_amdgcn_wmma_f32_16x16x4_f32(false, a, false, b, (short)0, acc, false, false);
    }
#pragma unroll
    for (int r = 0; r < 8; r++) {
        int m = dm0 + r + half * 8;     // d index
        if (m < KPAD) E2T[(size_t)((kn0 + l15) * 3 + q) * KPAD + m] = acc[r];
    }
}

// ---------------------------------------------------------------------------
// Per-batch: euler pose, Rodrigues, kinematic chain -> A matrices + feat145.
// One thread per batch element; all state in registers.
// ---------------------------------------------------------------------------
__global__ void k_feat(const float* __restrict__ beta, const float* __restrict__ theta,
                       const float* __restrict__ wrist, const float* __restrict__ hands_comp,
                       const float* __restrict__ hands_mean, const float* __restrict__ SJ,
                       const float* __restrict__ Jt, float* __restrict__ featPad,
                       float* __restrict__ A_ws, int B) {
    int b = blockIdx.x * blockDim.x + threadIdx.x;
    if (b >= B) return;
    float bt[10], th[10];
#pragma unroll
    for (int i = 0; i < 10; i++) { bt[i] = beta[b * 10 + i]; th[i] = theta[b * 10 + i]; }

    float* feat = featPad + (size_t)b * KPAD;
#pragma unroll
    for (int i = 0; i < 10; i++) feat[i] = bt[i];
    feat[145] = 0.0f; feat[146] = 0.0f; feat[147] = 0.0f;

    float J0[3];
#pragma unroll
    for (int c = 0; c < 3; c++) {
        float acc = Jt[c];
        for (int d = 0; d < 10; d++) acc += bt[d] * SJ[d * 48 + c];
        J0[c] = acc;
    }

    float R0[9];
    rodrigues9(wrist[b * 3 + 0], wrist[b * 3 + 1], wrist[b * 3 + 2], R0);

    float* A = A_ws + (size_t)b * 192;
#pragma unroll
    for (int c = 0; c < 3; c++) {
        A[c * 4 + 0] = R0[c * 3 + 0];
        A[c * 4 + 1] = R0[c * 3 + 1];
        A[c * 4 + 2] = R0[c * 3 + 2];
        A[c * 4 + 3] = J0[c] - (R0[c * 3] * J0[0] + R0[c * 3 + 1] * J0[1] + R0[c * 3 + 2] * J0[2]);
    }

    for (int ch = 0; ch < 5; ch++) {
        float Gr[9], Gt[3], Jp[3];
#pragma unroll
        for (int i = 0; i < 9; i++) Gr[i] = R0[i];
#pragma unroll
        for (int c = 0; c < 3; c++) { Gt[c] = J0[c]; Jp[c] = J0[c]; }
        for (int s = 0; s < 3; s++) {
            int i = 1 + ch * 3 + s;       // joint 1..15
            int e = (i - 1) * 3;
            float r[3];
#pragma unroll
            for (int c = 0; c < 3; c++) {
                float acc = hands_mean[e + c];
                for (int d = 0; d < 10; d++) acc += th[d] * hands_comp[d * 45 + e + c];
                r[c] = acc;
            }
            float R[9];
            rodrigues9(r[0], r[1], r[2], R);
#pragma unroll
            for (int u = 0; u < 9; u++)
                feat[10 + (i - 1) * 9 + u] = R[u] - ((u % 4 == 0) ? 1.0f : 0.0f);

            float Ji[3];
#pragma unroll
            for (int c = 0; c < 3; c++) {
                float acc = Jt[i * 3 + c];
                for (int d = 0; d < 10; d++) acc += bt[d] * SJ[d * 48 + i * 3 + c];
                Ji[c] = acc;
            }
            float t0 = Ji[0] - Jp[0], t1 = Ji[1] - Jp[1], t2 = Ji[2] - Jp[2];
            float nGr[9], nGt[3];
#pragma unroll
            for (int rr = 0; rr < 3; rr++) {
#pragma unroll
                for (int cc = 0; cc < 3; cc++)
                    nGr[rr * 3 + cc] = Gr[rr * 3 + 0] * R[0 * 3 + cc] + Gr[rr * 3 + 1] * R[1 * 3 + cc] +
                                       Gr[rr * 3 + 2] * R[2 * 3 + cc];
                nGt[rr] = Gr[rr * 3 + 0] * t0 + Gr[rr * 3 + 1] * t1 + Gr[rr * 3 + 2] * t2 + Gt[rr];
            }
#pragma unroll
            for (int c = 0; c < 3; c++) {
                A[i * 12 + c * 4 + 0] = nGr[c * 3 + 0];
                A[i * 12 + c * 4 + 1] = nGr[c * 3 + 1];
                A[i * 12 + c * 4 + 2] = nGr[c * 3 + 2];
                A[i * 12 + c * 4 + 3] =
                    nGt[c] - (nGr[c * 3] * Ji[0] + nGr[c * 3 + 1] * Ji[1] + nGr[c * 3 + 2] * Ji[2]);
            }
#pragma unroll
            for (int u = 0; u < 9; u++) Gr[u] = nGr[u];
#pragma unroll
            for (int c = 0; c < 3; c++) { Gt[c] = nGt[c]; Jp[c] = Ji[c]; }
        }
    }
}

// ---------------------------------------------------------------------------
// Main GEMM: Mout[b][col] = feat145[b] . E2T[col][:] + biasC[col]
// [Bpad,148] x [148,1024]; one wave per 16(M)x64(N) strip, 4 accumulators.
// Branch-free: 5 b64 loads + 4 WMMA per K-step. grid (Bpad/16, 16), block 32.
// ---------------------------------------------------------------------------
__global__ void k_gemm(const float* __restrict__ featPad, const float* __restrict__ E2T,
                       const float* __restrict__ biasC, float* __restrict__ Mout) {
    const int lane = threadIdx.x;
    const int half = lane >> 4, l15 = lane & 15;
    const int bm0 = blockIdx.x * 16;
    const int n0  = blockIdx.y * 64;

    const float* ap  = featPad + (size_t)(bm0 + l15) * KPAD;
    const float* bp0 = E2T + (size_t)(n0 +  0 + l15) * KPAD;
    const float* bp1 = E2T + (size_t)(n0 + 16 + l15) * KPAD;
    const float* bp2 = E2T + (size_t)(n0 + 32 + l15) * KPAD;
    const float* bp3 = E2T + (size_t)(n0 + 48 + l15) * KPAD;

    v8f acc0 = {0.f,0.f,0.f,0.f,0.f,0.f,0.f,0.f};
    v8f acc1 = acc0, acc2 = acc0, acc3 = acc0;

    for (int kk = 0; kk < KPAD / 4; kk++) {
        int kb = kk * 4 + half * 2;
        v2f a  = *(const v2f*)(ap  + kb);
        v2f b0 = *(const v2f*)(bp0 + kb);
        v2f b1 = *(const v2f*)(bp1 + kb);
        v2f b2 = *(const v2f*)(bp2 + kb);
        v2f b3 = *(const v2f*)(bp3 + kb);
        acc0 = __builtin_amdgcn_wmma_f32_16x16x4_f32(false, a, false, b0, (short)0, acc0, false, false);
        acc1 = __builtin_amdgcn_wmma_f32_16x16x4_f32(false, a, false, b1, (short)0, acc1, false, false);
        acc2 = __builtin_amdgcn_wmma_f32_16x16x4_f32(false, a, false, b2, (short)0, acc2, false, false);
        acc3 = __builtin_amdgcn_wmma_f32_16x16x4_f32(false, a, false, b3, (short)0, acc3, false, false);
    }

    v8f accs[4] = {acc0, acc1, acc2, acc3};
#pragma unroll
    for (int t = 0; t < 4; t++) {
        int col = n0 + t * 16 + l15;
        float bc = biasC[col];
#pragma unroll
        for (int r = 0; r < 8; r++) {
            int m = bm0 + r + half * 8;
            Mout[(size_t)m * NCP + col] = accs[t][r] + bc;
        }
    }
}

// ---------------------------------------------------------------------------
// joints[b,jj,c] = sum_j ( A[b,j,c,0..2] . M[b,jj,j,0..2] + A[b,j,c,3]*M3[jj,j] )
// ---------------------------------------------------------------------------
__global__ void k_joints(const float* __restrict__ Mout, const float* __restrict__ M3,
                         const float* __restrict__ A_ws, float* __restrict__ out, int B) {
    int idx = blockIdx.x * blockDim.x + threadIdx.x;
    if (idx >= B * NJ_ALL) return;
    int b = idx / NJ_ALL, jj = idx % NJ_ALL;
    const float* A = A_ws + (size_t)b * 192;
    const float* M = Mout + (size_t)b * NCP + jj * 48;   // col = jj*48 + j*3 + q
    const float* m3 = M3 + jj * 16;
    float o0 = 0.f, o1 = 0.f, o2 = 0.f;
#pragma unroll
    for (int j = 0; j < 16; j++) {
        float m0 = M[j * 3 + 0], m1 = M[j * 3 + 1], m2 = M[j * 3 + 2], mw = m3[j];
        const float* Aj = A + j * 12;
        o0 += Aj[0] * m0 + Aj[1]  * m1 + Aj[2]  * m2 + Aj[3]  * mw;
        o1 += Aj[4] * m0 + Aj[5]  * m1 + Aj[6]  * m2 + Aj[7]  * mw;
        o2 += Aj[8] * m0 + Aj[9]  * m1 + Aj[10] * m2 + Aj[11] * mw;
    }
    out[idx * 3 + 0] = o0;
    out[idx * 3 + 1] = o1;
    out[idx * 3 + 2] = o2;
}

// ---------------------------------------------------------------------------
extern "C" void kernel_launch(void* const* d_in, const int* in_sizes, int n_in,
                              void* d_out, int out_size, void* d_ws, size_t ws_size,
                              hipStream_t stream) {
    const float* beta       = (const float*)d_in[0];
    const float* theta      = (const float*)d_in[1];
    const float* wrist      = (const float*)d_in[2];
    const float* v_template = (const float*)d_in[3];
    const float* shapedirs  = (const float*)d_in[4];
    const float* posedirs   = (const float*)d_in[5];
    const float* Jreg       = (const float*)d_in[6];
    const float* hands_comp = (const float*)d_in[7];
    const float* hands_mean = (const float*)d_in[8];
    const float* weights    = (const float*)d_in[9];
    const int B    = in_sizes[0] / 10;
    const int Bpad = ((B + 15) / 16) * 16;

    float* ws = (float*)d_ws;
    float* SJ      = ws; ws += 10 * 48;
    float* Jt      = ws; ws += 48;
    float* W2T     = ws; ws += (size_t)NK2 * VPAD;
    float* DqPad   = ws; ws += (size_t)3 * DROWS * VPAD;
    float* E2T     = ws; ws += (size_t)NCP * KPAD;
    float* biasC   = ws; ws += NCP;
    float* M3      = ws; ws += NK2;
    float* featPad = ws; ws += (size_t)Bpad * KPAD;
    float* A_ws    = ws; ws += (size_t)B * 192;
    float* Mout    = ws;                       // Bpad * NCP

    // Batch-independent staging (all zero-padded / transposed for WMMA)
    k_SJ<<<(528 + 127) / 128, 128, 0, stream>>>(shapedirs, v_template, Jreg, SJ, Jt);
    k_W2T<<<(NK2 * VPAD + 255) / 256, 256, 0, stream>>>(Jreg, weights, W2T);
    k_Dq<<<(3 * DROWS * VPAD + 255) / 256, 256, 0, stream>>>(shapedirs, posedirs, DqPad);
    k_zero<<<(NCP * KPAD + 255) / 256, 256, 0, stream>>>(E2T, NCP * KPAD);
    k_bias<<<(NCP + NK2 + 127) / 128, 128, 0, stream>>>(W2T, v_template, biasC, M3);
    k_E2<<<dim3(10, 21, 3), 32, 0, stream>>>(DqPad, W2T, E2T);

    // Per-batch small work (Rodrigues + kinematic chain + features)
    if (Bpad > B)
        k_zero<<<((Bpad - B) * KPAD + 255) / 256, 256, 0, stream>>>(featPad + (size_t)B * KPAD,
                                                                    (Bpad - B) * KPAD);
    k_feat<<<(B + 127) / 128, 128, 0, stream>>>(beta, theta, wrist, hands_comp, hands_mean,
                                                SJ, Jt, featPad, A_ws, B);

    // Main WMMA GEMM: M = feat @ E2 + bias   ([Bpad,148] x [148,1024])
    k_gemm<<<dim3(Bpad / 16, NCP / 64), 32, 0, stream>>>(featPad, E2T, biasC, Mout);

    // Epilogue: contract with per-batch A matrices -> joints [B,21,3]
    k_joints<<<(B * NJ_ALL + 255) / 256, 256, 0, stream>>>(Mout, M3, A_ws, (float*)d_out, B);
}